// SSDLoss_6047313952802
// MI455X (gfx1250) — compile-verified
//
#include <hip/hip_runtime.h>
#include <hip/hip_bf16.h>
#include <math.h>
#include <stdint.h>

// ---- problem constants (match reference) ----
#define BB   32
#define AA   49152
#define MM   64
#define TPB  256
#define NBLK (AA / TPB)   // 192

#define ALPHA        0.25f
#define POS_TH       0.5f
#define NEG_TH       0.3f
#define CENTER_VAR   0.1f
#define SIZE_VAR     0.2f

typedef __attribute__((ext_vector_type(2))) float v2f;
typedef __attribute__((ext_vector_type(8))) float v8f;

// ---------------- workspace layout (bytes) ----------------
// acc_f : 4*BB floats  : [0,BB)=pos_sum [BB,2BB)=cls_sum [2BB,3BB)=reg_sum [3BB,4BB)=sum_gt
// acc_i : 7*BB ints    : [0,BB)=npos [BB)=nneg [2BB)=n_gt [3BB)=k [4BB)=k_remain [5BB)=prefix(bits) [6BB)=cnt_gt
// g_hist: BB*256 uints
// neg_loss: BB*AA floats
#define OFF_ACCF 0
#define OFF_ACCI (OFF_ACCF + 4 * BB * 4)
#define OFF_HIST ((OFF_ACCI + 7 * BB * 4 + 255) & ~255)
#define OFF_NEGL ((OFF_HIST + BB * 256 * 4 + 255) & ~255)

__device__ __forceinline__ void wait_asynccnt0() {
#if __has_builtin(__builtin_amdgcn_s_wait_asynccnt)
    __builtin_amdgcn_s_wait_asynccnt(0);
#else
    asm volatile("s_wait_asynccnt 0x0" ::: "memory");
#endif
}

__device__ __forceinline__ float block_reduce_sum(float v, float* buf) {
    const int tid = threadIdx.x;
    buf[tid] = v;
    __syncthreads();
    for (int s = TPB / 2; s > 0; s >>= 1) {
        if (tid < s) buf[tid] += buf[tid + s];
        __syncthreads();
    }
    float r = buf[0];
    __syncthreads();
    return r;
}

// ---------------- kernel 0: zero accumulators + histograms ----------------
__global__ void k_init(float* acc_f, int* acc_i, unsigned* g_hist) {
    int t = blockIdx.x * blockDim.x + threadIdx.x;   // grid covers BB*256
    if (t < BB * 256) g_hist[t] = 0u;
    if (t < BB) {
        acc_f[t] = 0.0f; acc_f[BB + t] = 0.0f; acc_f[2 * BB + t] = 0.0f; acc_f[3 * BB + t] = 0.0f;
        for (int j = 0; j < 7; ++j) acc_i[j * BB + t] = 0;
    }
}

// ---------------- kernel 1: match + per-anchor losses + partial sums ----------------
__global__ void __launch_bounds__(TPB)
k_match(const float* __restrict__ cls_p, const float* __restrict__ reg_p,
        const float* __restrict__ gt, const int* __restrict__ valid,
        const float* __restrict__ anchors,
        float* __restrict__ neg_loss, float* acc_f, int* acc_i) {
    // raw async-staged GT data (memory -> LDS, no VGPR round trip)
    __shared__ float4 s_graw[MM];
    __shared__ int    s_vraw[MM];
    // derived per-GT quantities
    __shared__ float s_gx1[MM], s_gy1[MM], s_gx2[MM], s_gy2[MM], s_area[MM];
    __shared__ float s_cx[MM], s_cy[MM], s_gw[MM], s_gh[MM];
    __shared__ int   s_val[MM];
    __shared__ float red[TPB];

    const int tid = threadIdx.x;
    const int b   = blockIdx.y;
    const int a   = blockIdx.x * TPB + tid;

    // gfx1250 async copies: GT boxes (16B/lane) + valid mask (4B/lane) into LDS.
    if (tid < MM) {
        uint32_t lds_g = (uint32_t)(uintptr_t)(&s_graw[tid]);
        uint64_t ga    = (uint64_t)(uintptr_t)(gt + ((size_t)b * MM + tid) * 4);
        asm volatile("global_load_async_to_lds_b128 %0, %1, off"
                     :: "v"(lds_g), "v"(ga) : "memory");
        uint32_t lds_v = (uint32_t)(uintptr_t)(&s_vraw[tid]);
        uint64_t va    = (uint64_t)(uintptr_t)(valid + (size_t)b * MM + tid);
        asm volatile("global_load_async_to_lds_b32 %0, %1, off"
                     :: "v"(lds_v), "v"(va) : "memory");
    }

    // gfx1250 prefetch of the prediction rows this block will read
    __builtin_prefetch(cls_p + ((size_t)b * AA + a) * 2, 0, 1);
    __builtin_prefetch(reg_p + ((size_t)b * AA + a) * 4, 0, 1);

    wait_asynccnt0();
    __syncthreads();

    if (tid < MM) {
        float4 g = s_graw[tid];
        s_cx[tid] = g.x; s_cy[tid] = g.y; s_gw[tid] = g.z; s_gh[tid] = g.w;
        s_gx1[tid] = g.x - 0.5f * g.z; s_gy1[tid] = g.y - 0.5f * g.w;
        s_gx2[tid] = g.x + 0.5f * g.z; s_gy2[tid] = g.y + 0.5f * g.w;
        s_area[tid] = g.z * g.w;
        s_val[tid] = s_vraw[tid];
    }
    __syncthreads();

    float4 an = ((const float4*)anchors)[a];
    const float ax1 = an.x - 0.5f * an.z, ay1 = an.y - 0.5f * an.w;
    const float ax2 = an.x + 0.5f * an.z, ay2 = an.y + 0.5f * an.w;
    const float area_a = an.z * an.w;

    float best = -1e30f;
    int   bi   = 0;
    #pragma unroll 8
    for (int m = 0; m < MM; ++m) {
        float ix1 = fmaxf(ax1, s_gx1[m]), iy1 = fmaxf(ay1, s_gy1[m]);
        float ix2 = fminf(ax2, s_gx2[m]), iy2 = fminf(ay2, s_gy2[m]);
        float iw = fmaxf(ix2 - ix1, 0.0f), ih = fmaxf(iy2 - iy1, 0.0f);
        float inter = iw * ih;
        float uni = area_a + s_area[m] - inter;
        float iou = inter / fmaxf(uni, 1e-6f);
        float v = (s_val[m] > 0) ? iou : -1.0f;
        if (v > best) { best = v; bi = m; }
    }

    const bool pos = best > POS_TH;
    const bool neg = best < NEG_TH;

    // focal loss, 2 classes
    float2 lg = ((const float2*)cls_p)[(size_t)b * AA + a];
    float l0 = lg.x, l1 = lg.y;
    float mx = fmaxf(l0, l1);
    float lse = mx + logf(expf(l0 - mx) + expf(l1 - mx));
    float lt = pos ? l1 : l0;
    float ce = lse - lt;
    float pt = expf(-ce);
    float at = pos ? ALPHA : (1.0f - ALPHA);
    float om = 1.0f - pt;
    float closs = at * om * om * ce;

    // regression (only matters where pos)
    float4 rp = ((const float4*)reg_p)[(size_t)b * AA + a];
    float tx = (s_cx[bi] - an.x) / (an.z * CENTER_VAR);
    float ty = (s_cy[bi] - an.y) / (an.w * CENTER_VAR);
    float tw = logf(fmaxf(s_gw[bi] / fmaxf(an.z, 1e-6f), 1e-6f)) / SIZE_VAR;
    float th = logf(fmaxf(s_gh[bi] / fmaxf(an.w, 1e-6f), 1e-6f)) / SIZE_VAR;
    float rl = 0.0f;
    {
        float d;
        d = fabsf(rp.x - tx); rl += (d < 1.0f) ? 0.5f * d * d : d - 0.5f;
        d = fabsf(rp.y - ty); rl += (d < 1.0f) ? 0.5f * d * d : d - 0.5f;
        d = fabsf(rp.z - tw); rl += (d < 1.0f) ? 0.5f * d * d : d - 0.5f;
        d = fabsf(rp.w - th); rl += (d < 1.0f) ? 0.5f * d * d : d - 0.5f;
    }

    neg_loss[(size_t)b * AA + a] = neg ? closs : 0.0f;

    float s_pos  = block_reduce_sum(pos ? closs : 0.0f, red);
    float s_cls  = block_reduce_sum(closs, red);
    float s_reg  = block_reduce_sum(pos ? rl : 0.0f, red);
    float s_npos = block_reduce_sum(pos ? 1.0f : 0.0f, red);
    float s_nneg = block_reduce_sum(neg ? 1.0f : 0.0f, red);

    if (tid == 0) {
        atomicAdd(acc_f + b, s_pos);
        atomicAdd(acc_f + BB + b, s_cls);
        atomicAdd(acc_f + 2 * BB + b, s_reg);
        atomicAdd(acc_i + b, (int)(s_npos + 0.5f));
        atomicAdd(acc_i + BB + b, (int)(s_nneg + 0.5f));
    }
}

// ---------------- kernel 2: per-image k + radix-select init ----------------
__global__ void k_select_init(const int* __restrict__ pos_mask, int* acc_i) {
    int b = blockIdx.x;
    if (threadIdx.x == 0) {
        int ngt = 0;
        for (int m = 0; m < MM; ++m) ngt += (pos_mask[b * MM + m] > 0) ? 1 : 0;
        int npos = acc_i[b];
        int nneg = acc_i[BB + b];
        int cap = 3 * ((npos > 1) ? npos : 1);
        int k = (nneg < cap) ? nneg : cap;
        acc_i[2 * BB + b] = ngt;
        acc_i[3 * BB + b] = k;
        acc_i[4 * BB + b] = k;      // k_remain
        acc_i[5 * BB + b] = 0;      // prefix bits
    }
}

// ---------------- kernel 3: histogram pass of radix select ----------------
__global__ void __launch_bounds__(TPB)
k_hist(const float* __restrict__ neg_loss, const int* acc_i, unsigned* g_hist, int p) {
    __shared__ unsigned h[256];
    const int tid = threadIdx.x;
    const int b = blockIdx.y;
    const int a = blockIdx.x * TPB + tid;
    h[tid] = 0u;
    __syncthreads();

    unsigned bits = __float_as_uint(neg_loss[(size_t)b * AA + a]);
    unsigned prefix = (unsigned)acc_i[5 * BB + b];
    unsigned mask = (p == 0) ? 0u : (~0u << (32 - 8 * p));
    int shift = 24 - 8 * p;
    if (((bits ^ prefix) & mask) == 0u)
        atomicAdd(&h[(bits >> shift) & 0xFFu], 1u);
    __syncthreads();
    if (h[tid]) atomicAdd(&g_hist[b * 256 + tid], h[tid]);
}

// ---------------- kernel 4: pick radix bucket, update prefix/k_remain ----------------
__global__ void k_select(unsigned* g_hist, int* acc_i, int p) {
    __shared__ unsigned h[256];
    const int tid = threadIdx.x;
    const int b = blockIdx.x;
    h[tid] = g_hist[b * 256 + tid];
    g_hist[b * 256 + tid] = 0u;       // ready for next pass
    __syncthreads();
    if (tid == 0) {
        int kr = acc_i[4 * BB + b];
        if (kr > 0) {
            int acc = 0, j;
            for (j = 255; j >= 0; --j) {
                int c = (int)h[j];
                if (acc + c >= kr) break;
                acc += c;
            }
            if (j < 0) j = 0;
            int shift = 24 - 8 * p;
            unsigned prefix = (unsigned)acc_i[5 * BB + b];
            prefix |= ((unsigned)j) << shift;
            acc_i[5 * BB + b] = (int)prefix;
            acc_i[4 * BB + b] = kr - acc;
        }
    }
}

// ---------------- kernel 5: sum/count of values strictly above threshold ----------------
__global__ void __launch_bounds__(TPB)
k_topsum(const float* __restrict__ neg_loss, const int* acc_i, float* acc_f, int* acc_i_out) {
    __shared__ float red[TPB];
    const int tid = threadIdx.x;
    const int b = blockIdx.y;
    const int a = blockIdx.x * TPB + tid;
    unsigned Tb = (unsigned)acc_i[5 * BB + b];
    float v = neg_loss[(size_t)b * AA + a];
    bool gt = __float_as_uint(v) > Tb;
    float s = block_reduce_sum(gt ? v : 0.0f, red);
    float c = block_reduce_sum(gt ? 1.0f : 0.0f, red);
    if (tid == 0) {
        atomicAdd(acc_f + 3 * BB + b, s);
        atomicAdd(acc_i_out + 6 * BB + b, (int)(c + 0.5f));
    }
}

// ---------------- WMMA full-wave (32-lane) sum via 16x16x4 f32 ----------------
// Stage 1: A = {x,0} (A[m][0]=x_m, A[m][2]=x_{m+16}), B = ones -> D1[m][n] = x_m + x_{m+16}.
// Stage 2: 4 accumulating WMMAs feeding the D1 C-layout VGPR pairs back as A
// (rows are constant across n) sum the 16 row-sums -> total broadcast to all lanes.
__device__ __forceinline__ float wmma_sum32(float x) {
    v2f a;    a[0] = x;    a[1] = 0.0f;
    v2f ones; ones[0] = 1.0f; ones[1] = 1.0f;
    v8f c0 = {};
    v8f d1 = __builtin_amdgcn_wmma_f32_16x16x4_f32(false, a, false, ones, (short)0, c0, false, false);
    v8f acc = {};
    v2f p;
    p[0] = d1[0]; p[1] = d1[1];
    acc = __builtin_amdgcn_wmma_f32_16x16x4_f32(false, p, false, ones, (short)0, acc, false, false);
    p[0] = d1[2]; p[1] = d1[3];
    acc = __builtin_amdgcn_wmma_f32_16x16x4_f32(false, p, false, ones, (short)0, acc, false, false);
    p[0] = d1[4]; p[1] = d1[5];
    acc = __builtin_amdgcn_wmma_f32_16x16x4_f32(false, p, false, ones, (short)0, acc, false, false);
    p[0] = d1[6]; p[1] = d1[7];
    acc = __builtin_amdgcn_wmma_f32_16x16x4_f32(false, p, false, ones, (short)0, acc, false, false);
    return acc[0];
}

// ---------------- kernel 6: per-image combine + WMMA batch mean ----------------
__global__ void k_finalize(const float* acc_f, const int* acc_i, float* out) {
    const int b = threadIdx.x;           // 32 threads = one full wave, lane == image
    float pos_sum = acc_f[b];
    float cls_sum = acc_f[BB + b];
    float reg_sum = acc_f[2 * BB + b];
    float sum_gt  = acc_f[3 * BB + b];
    int npos   = acc_i[b];
    int ngt    = acc_i[2 * BB + b];
    int k      = acc_i[3 * BB + b];
    int cnt_gt = acc_i[6 * BB + b];
    float T = __uint_as_float((unsigned)acc_i[5 * BB + b]);

    float neg_sum = (k > 0) ? (sum_gt + (float)(k - cnt_gt) * T) : 0.0f;
    float npos_f = (float)((npos > 1) ? npos : 1);
    float cls_with = (pos_sum + neg_sum) / npos_f;
    float cls_no   = (k > 0) ? (neg_sum / (float)((k > 1) ? k : 1)) : 0.0f;
    float cls_mined = (npos > 0) ? cls_with : cls_no;
    float cls_b = (ngt == 0) ? (cls_sum / (float)AA) : cls_mined;
    float reg_b = reg_sum / npos_f;

    float tc = wmma_sum32(cls_b) * (1.0f / (float)BB);
    float tr = wmma_sum32(reg_b) * (1.0f / (float)BB);
    if (b == 0) {
        out[0] = tc + tr;   // total (REG_WEIGHT = 1)
        out[1] = tc;        // total_cls
        out[2] = tr;        // total_reg
    }
}

extern "C" void kernel_launch(void* const* d_in, const int* in_sizes, int n_in,
                              void* d_out, int out_size, void* d_ws, size_t ws_size,
                              hipStream_t stream) {
    const float* cls_p   = (const float*)d_in[0];
    const float* reg_p   = (const float*)d_in[1];
    // d_in[2] = cls_targets (unused by the reference loss math)
    const float* gt      = (const float*)d_in[3];
    const int*   pmask   = (const int*)d_in[4];
    const float* anchors = (const float*)d_in[5];
    float* out = (float*)d_out;

    char* ws = (char*)d_ws;
    float*    acc_f    = (float*)(ws + OFF_ACCF);
    int*      acc_i    = (int*)(ws + OFF_ACCI);
    unsigned* g_hist   = (unsigned*)(ws + OFF_HIST);
    float*    neg_loss = (float*)(ws + OFF_NEGL);

    dim3 gridA(NBLK, BB);

    k_init<<<BB, 256, 0, stream>>>(acc_f, acc_i, g_hist);
    k_match<<<gridA, TPB, 0, stream>>>(cls_p, reg_p, gt, pmask, anchors,
                                       neg_loss, acc_f, acc_i);
    k_select_init<<<BB, 32, 0, stream>>>(pmask, acc_i);
    for (int p = 0; p < 4; ++p) {
        k_hist<<<gridA, TPB, 0, stream>>>(neg_loss, acc_i, g_hist, p);
        k_select<<<BB, 256, 0, stream>>>(g_hist, acc_i, p);
    }
    k_topsum<<<gridA, TPB, 0, stream>>>(neg_loss, acc_i, acc_f, acc_i);
    k_finalize<<<1, 32, 0, stream>>>(acc_f, acc_i, out);
}